// RotatedIOULoss_61151744360583
// MI455X (gfx1250) — compile-verified
//
#include <hip/hip_runtime.h>
#include <math.h>

typedef float v2f __attribute__((ext_vector_type(2)));
typedef float v8f __attribute__((ext_vector_type(8)));

#define BLOCK 256

// Wave32 reduction: one V_WMMA_F32_16X16X4_F32 folds 32 lane partials to 16
// pair-sums (A = ones 16x4, B rows0/1 = lanes 0-15/16-31), then 4 immediate
// ds_swizzle XOR butterflies. EXEC must be all-ones here (straight-line code).
__device__ __forceinline__ float wave_sum_wmma(float p) {
    v2f a; a[0] = 1.0f; a[1] = 1.0f;     // A: 16x4 all-ones
    v2f b; b[0] = p;    b[1] = 0.0f;     // B: 4x16, rows 0/1 = partials, rows 2/3 = 0
    v8f c = {};
    v8f d = __builtin_amdgcn_wmma_f32_16x16x4_f32(
        /*neg_a=*/false, a, /*neg_b=*/false, b,
        /*c_mod=*/(short)0, c, /*reuse_a=*/false, /*reuse_b=*/false);
    float s = d[0];                      // lane i: p[i%16] + p[i%16+16]
    // group-of-32 ds_swizzle: offset = xor<<10 | or<<5 | and(0x1f)
    s += __int_as_float(__builtin_amdgcn_ds_swizzle(__float_as_int(s), 0x041F)); // XOR 1
    s += __int_as_float(__builtin_amdgcn_ds_swizzle(__float_as_int(s), 0x081F)); // XOR 2
    s += __int_as_float(__builtin_amdgcn_ds_swizzle(__float_as_int(s), 0x101F)); // XOR 4
    s += __int_as_float(__builtin_amdgcn_ds_swizzle(__float_as_int(s), 0x201F)); // XOR 8
    return s;                            // every lane holds the wave total
}

// Liang-Barsky clip of segment p->q against a CCW convex quad (corners rx/ry,
// edge vectors ex/ey). Returns cross(P,Q) of the clipped sub-segment (its
// shoelace contribution to 2*Area of the intersection polygon), or 0 if the
// segment misses the quad. Interval logic:
//   per half-plane k, inside is f = cross(e_k, p - r_k) >= 0; the in-interval
//   along the segment is [ts,1], [0,ts], [0,1] or empty; t0/t1 accumulate the
//   intersection of all four. Empty is forced via t0 = 2 (> any t1).
__device__ __forceinline__ float clip_cross(
    float px, float py, float qx, float qy,
    const float rx[4], const float ry[4],
    const float ex[4], const float ey[4]) {
    float t0 = 0.0f, t1 = 1.0f;
#pragma unroll
    for (int k = 0; k < 4; ++k) {
        float fp = ex[k] * (py - ry[k]) - ey[k] * (px - rx[k]);
        float fq = ex[k] * (qy - ry[k]) - ey[k] * (qx - rx[k]);
        bool pin = fp >= 0.0f, qin = fq >= 0.0f;
        float ts = fp / (fp - fq);       // only used when pin != qin (finite, in [0,1])
        t0 = (!pin && !qin) ? 2.0f : ((!pin) ? fmaxf(t0, ts) : t0);
        t1 = (pin && !qin) ? fminf(t1, ts) : t1;
    }
    float dx = qx - px, dy = qy - py;
    float Px = fmaf(t0, dx, px), Py = fmaf(t0, dy, py);
    float Qx = fmaf(t1, dx, px), Qy = fmaf(t1, dy, py);
    float c = Px * Qy - Py * Qx;
    return (t0 < t1) ? c : 0.0f;
}

__global__ __launch_bounds__(BLOCK) void riou_main(
    const float* __restrict__ pred, const float* __restrict__ target,
    const float* __restrict__ weight, float* __restrict__ block_sums, int N) {

    const int tid     = blockIdx.x * BLOCK + threadIdx.x;
    const bool active = tid < N;
    const int idx     = active ? tid : (N - 1);   // clamp: keep EXEC uniform

    float pd[7], td[7];
#pragma unroll
    for (int k = 0; k < 7; ++k) pd[k] = pred[idx * 7 + k];
#pragma unroll
    for (int k = 0; k < 7; ++k) td[k] = target[idx * 7 + k];
    const float wt = weight[idx];

    // ---- FCOS OBB decode (loc = 0) ----
    float w1 = pd[0] + pd[3], l1 = pd[1] + pd[4], h1 = pd[2] + pd[5];
    float ox1 = (pd[3] - pd[0]) * 0.5f, oy1 = (pd[4] - pd[1]) * 0.5f, oz1 = (pd[5] - pd[2]) * 0.5f;
    float s1, c1; sincosf(pd[6], &s1, &c1);
    float x1 = ox1 * c1 - oy1 * s1;
    float y1 = ox1 * s1 + oy1 * c1;

    float w2 = td[0] + td[3], l2 = td[1] + td[4], h2 = td[2] + td[5];
    float ox2 = (td[3] - td[0]) * 0.5f, oy2 = (td[4] - td[1]) * 0.5f, oz2 = (td[5] - td[2]) * 0.5f;
    float s2, c2; sincosf(td[6], &s2, &c2);
    float x2c = ox2 * c2 - oy2 * s2;
    float y2c = ox2 * s2 + oy2 * c2;

    // ---- corners (CCW for w,l > 0) and edge vectors ----
    const float SX[4] = {0.5f, -0.5f, -0.5f, 0.5f};
    const float SY[4] = {0.5f, 0.5f, -0.5f, -0.5f};
    float ax[4], ay[4], bx[4], by[4];
#pragma unroll
    for (int k = 0; k < 4; ++k) {
        float qx = SX[k] * w1, qy = SY[k] * l1;
        ax[k] = qx * c1 - qy * s1 + x1;
        ay[k] = qx * s1 + qy * c1 + y1;
        float rx = SX[k] * w2, ry = SY[k] * l2;
        bx[k] = rx * c2 - ry * s2 + x2c;
        by[k] = rx * s2 + ry * c2 + y2c;
    }
    float eax[4], eay[4], ebx[4], eby[4];
#pragma unroll
    for (int k = 0; k < 4; ++k) {
        eax[k] = ax[(k + 1) & 3] - ax[k];
        eay[k] = ay[(k + 1) & 3] - ay[k];
        ebx[k] = bx[(k + 1) & 3] - bx[k];
        eby[k] = by[(k + 1) & 3] - by[k];
    }

    // ---- 2*signed area of A (intersect) B via clipped-edge shoelace:
    //      arcs of A-edges inside B plus arcs of B-edges inside A; shared
    //      arc endpoints make all inter-arc cross terms vanish. ----
    float ssum = 0.0f;
#pragma unroll
    for (int i = 0; i < 4; ++i)
        ssum += clip_cross(ax[i], ay[i], ax[(i + 1) & 3], ay[(i + 1) & 3],
                           bx, by, ebx, eby);
#pragma unroll
    for (int j = 0; j < 4; ++j)
        ssum += clip_cross(bx[j], by[j], bx[(j + 1) & 3], by[(j + 1) & 3],
                           ax, ay, eax, eay);
    float inter = 0.5f * fabsf(ssum);

    // ---- 3D IoU + loss ----
    float area1 = w1 * l1, area2 = w2 * l2;
    float u2 = area1 + area2 - inter;
    float iou2d = inter / u2;
    float zmax = fminf(oz1 + h1 * 0.5f, oz2 + h2 * 0.5f);
    float zmin = fmaxf(oz1 - h1 * 0.5f, oz2 - h2 * 0.5f);
    float zo = fmaxf(zmax - zmin, 0.0f);
    float inter3 = iou2d * u2 * zo;
    float u3 = area1 * h1 + area2 * h2 - inter3;
    float loss = -logf((inter3 + 1.0f) / (u3 + 1.0f));
    float part = active ? loss * wt : 0.0f;

    // ---- reduction: WMMA wave sum -> LDS -> fixed-order block sum ----
    float wsum = wave_sum_wmma(part);

    __shared__ float lsum[BLOCK / 32];
    int lane = threadIdx.x & 31;
    int wv   = threadIdx.x >> 5;
    if (lane == 0) lsum[wv] = wsum;
    __syncthreads();
    if (threadIdx.x == 0) {
        float b = 0.0f;
#pragma unroll
        for (int k = 0; k < BLOCK / 32; ++k) b += lsum[k];
        block_sums[blockIdx.x] = b;
    }
}

__global__ __launch_bounds__(1024) void riou_final(
    const float* __restrict__ bs, float* __restrict__ out, int nb) {
    __shared__ float l[1024];
    int t = threadIdx.x;
    float a = 0.0f;
    for (int i = t; i < nb; i += 1024) a += bs[i];   // fixed order per slot
    l[t] = a;
    __syncthreads();
#pragma unroll
    for (int st = 512; st > 0; st >>= 1) {
        if (t < st) l[t] += l[t + st];
        __syncthreads();
    }
    if (t == 0) out[0] = l[0];
}

extern "C" void kernel_launch(void* const* d_in, const int* in_sizes, int n_in,
                              void* d_out, int out_size, void* d_ws, size_t ws_size,
                              hipStream_t stream) {
    const float* pred   = (const float*)d_in[0];
    const float* target = (const float*)d_in[1];
    const float* weight = (const float*)d_in[2];
    const int N = in_sizes[2];                 // weight has N elements
    float* ws = (float*)d_ws;
    const int nb = (N + BLOCK - 1) / BLOCK;    // 977 blocks for N=250000
    riou_main<<<nb, BLOCK, 0, stream>>>(pred, target, weight, ws, N);
    riou_final<<<1, 1024, 0, stream>>>(ws, (float*)d_out, nb);
}